// DropBlockLIF_44143673868824
// MI455X (gfx1250) — compile-verified
//
#include <hip/hip_runtime.h>
#include <float.h>

typedef float v4f __attribute__((ext_vector_type(4)));

#define T_STEPS 5
#define BS      32
#define NC      64
#define NH      64
#define NW      64
// floats per n: NC*NH*NW = 262144  -> 65536 v4f
#define N_VOL4  65536

#if defined(__AMDGCN__) && __has_builtin(__builtin_amdgcn_global_load_async_to_lds_b32) && __has_builtin(__builtin_amdgcn_s_wait_asynccnt)
#define USE_ASYNC_LDS 1
#else
#define USE_ASYNC_LDS 0
#endif

__global__ __launch_bounds__(256) void dropblock_lif_kernel(
    const float* __restrict__ x,
    const float* __restrict__ mrand,
    float* __restrict__ out)
{
    const int h   = blockIdx.x;   // 0..63
    const int b   = blockIdx.y;   // 0..31
    const int tid = threadIdx.x;  // 0..255

    // LDS: 5 time-steps x 7 neighbor rows x 64 cols of mask_rand, then
    // separable-min scratch and the final per-(t,w) dropblock mask.
    __shared__ float rtile[T_STEPS * 7 * NW];        // 2240 floats
    __shared__ float rowm [T_STEPS * NW];            // 320 floats
    __shared__ v4f   bm_sv[T_STEPS * (NW / 4)];      // 320 floats (16B aligned)
    float* bm_s = (float*)bm_sv;

    // ---- Stage mask_rand neighborhood rows h-3..h+3 (clamped -> +inf) ----
    for (int i = tid; i < T_STEPS * 7 * NW; i += 256)
        rtile[i] = FLT_MAX;
    __syncthreads();

    for (int i = tid; i < T_STEPS * 7 * NW; i += 256) {
        const int t  = i / (7 * NW);
        const int r  = (i / NW) % 7;
        const int w  = i % NW;
        const int hr = h - 3 + r;
        if (hr >= 0 && hr < NH) {
            const float* g = mrand + ((size_t)(t * BS + b) * (NH * NW) + hr * NW + w);
#if USE_ASYNC_LDS
            __builtin_amdgcn_global_load_async_to_lds_b32(
                (__attribute__((address_space(1))) int*)const_cast<float*>(g),
                (__attribute__((address_space(3))) int*)&rtile[i],
                0, 0);
#else
            rtile[i] = *g;
#endif
        }
    }
#if USE_ASYNC_LDS
    __builtin_amdgcn_s_wait_asynccnt(0);
#endif
    __syncthreads();

    // ---- Separable 7x7 min-pool: min over the 7 rows, then w-window ----
    for (int i = tid; i < T_STEPS * NW; i += 256) {
        const int t = i >> 6;
        const int w = i & 63;
        const float* base = &rtile[t * (7 * NW) + w];
        float m = base[0];
        #pragma unroll
        for (int r = 1; r < 7; ++r) m = fminf(m, base[r * NW]);
        rowm[i] = m;
    }
    __syncthreads();

    const float gamma = (float)(0.1 / 49.0);  // DROP_RATE / BLOCK_SIZE^2, f64 -> f32 like jnp
    for (int i = tid; i < T_STEPS * NW; i += 256) {
        const int t  = i >> 6;
        const int w  = i & 63;
        const int w0 = (w - 3 > 0)  ? (w - 3) : 0;
        const int w1 = (w + 3 < 63) ? (w + 3) : 63;
        float m = rowm[t * NW + w0];
        for (int wc = w0 + 1; wc <= w1; ++wc) m = fminf(m, rowm[t * NW + wc]);
        // bm = 1 - maxpool(rand < gamma)  ==  (min_window >= gamma)
        bm_s[i] = (m >= gamma) ? 1.0f : 0.0f;
    }
    __syncthreads();

    // ---- LIF recurrence, streamed with b128 NT loads/stores ----
    const v4f* __restrict__ x4 = (const v4f*)x;
    v4f* __restrict__ o4 = (v4f*)out;

    #pragma unroll
    for (int k = 0; k < 4; ++k) {
        const int item = tid + k * 256;   // 0..1023  -> (c, w4)
        const int c    = item >> 4;       // 0..63
        const int w4   = item & 15;       // 0..15 (float4 column)
        const size_t base = (size_t)c * (NH * NW / 4) + (size_t)h * (NW / 4) + w4;

        v4f u = 0.0f;
        #pragma unroll
        for (int t = 0; t < T_STEPS; ++t) {
            const size_t idx = (size_t)(t * BS + b) * N_VOL4 + base;
            const v4f xt = __builtin_nontemporal_load(x4 + idx);

            // u = TAU*u*(1 - spike(u)) + xt   (TAU = 0.5, VTH = 1.0)
            u.x = ((u.x > 1.0f) ? 0.0f : 0.5f * u.x) + xt.x;
            u.y = ((u.y > 1.0f) ? 0.0f : 0.5f * u.y) + xt.y;
            u.z = ((u.z > 1.0f) ? 0.0f : 0.5f * u.z) + xt.z;
            u.w = ((u.w > 1.0f) ? 0.0f : 0.5f * u.w) + xt.w;

            const v4f bmv = bm_sv[t * (NW / 4) + w4];
            v4f o;
            o.x = (u.x > 1.0f) ? bmv.x : 0.0f;
            o.y = (u.y > 1.0f) ? bmv.y : 0.0f;
            o.z = (u.z > 1.0f) ? bmv.z : 0.0f;
            o.w = (u.w > 1.0f) ? bmv.w : 0.0f;

            __builtin_nontemporal_store(o, o4 + idx);
        }
    }
}

extern "C" void kernel_launch(void* const* d_in, const int* in_sizes, int n_in,
                              void* d_out, int out_size, void* d_ws, size_t ws_size,
                              hipStream_t stream) {
    (void)in_sizes; (void)n_in; (void)d_ws; (void)ws_size; (void)out_size;
    const float* x     = (const float*)d_in[0];   // [160, 64, 64, 64] f32
    const float* mrand = (const float*)d_in[1];   // [160, 64, 64]    f32
    float* out         = (float*)d_out;           // [160, 64, 64, 64] f32

    dim3 grid(NH, BS);   // (h, b) -> 2048 workgroups
    dropblock_lif_kernel<<<grid, 256, 0, stream>>>(x, mrand, out);
}